// HBVMulTDET_17437567222010
// MI455X (gfx1250) — compile-verified
//
#include <hip/hip_runtime.h>
#include <stdint.h>

// HBV hydrological scan on gfx1250 (MI455X).
// - 1 thread per (grid cell g, multiplier m); block = 32 g x 8 m = 256 thr = 8 wave32.
// - Forcing data [T,G,3] staged through LDS with double-buffered TDM
//   (tensor_load_to_lds) 2D tiles: rows = timesteps, cols = 32*3 floats.
// - pow via v_log_f32/v_exp_f32; outputs streamed with non-temporal stores.

#define TCH  32   // timesteps per LDS chunk
#define GPB  32   // grid cells per block
#define NMUL 8

typedef unsigned int u32x4 __attribute__((ext_vector_type(4)));
typedef int          i32x4 __attribute__((ext_vector_type(4)));
typedef int          i32x8 __attribute__((ext_vector_type(8)));

#if defined(__HIP_DEVICE_COMPILE__) && __has_builtin(__builtin_amdgcn_tensor_load_to_lds)
#define HAVE_TDM 1
#else
#define HAVE_TDM 0
#endif

// v_log_f32 computes log2(x); v_exp_f32 computes 2^x.
__device__ __forceinline__ float fast_log2(float x) {
#if __has_builtin(__builtin_amdgcn_logf)
  return __builtin_amdgcn_logf(x);
#else
  return log2f(x);
#endif
}
__device__ __forceinline__ float fast_exp2(float x) {
#if __has_builtin(__builtin_amdgcn_exp2f)
  return __builtin_amdgcn_exp2f(x);
#else
  return exp2f(x);
#endif
}

// Issue a TDM 2D tile load: copies rows x cols floats from global memory at gaddr (rows row_stride_elems apart) to LDS at lds_off, rows packed
// Descriptor bitfields per CDNA5 ISA sec 8.3/8.4 (D# group0 + group1), 2D tensor,
// data_size=4B, no padding/iterate/gather, workgroup_mask=0 (not in a cluster).
__device__ __forceinline__ void tdm_load_tile_2d(uint32_t lds_off, uint64_t gaddr,
                                                 uint32_t rows, uint32_t cols,
                                                 uint32_t row_stride_elems) {
#if HAVE_TDM
  u32x4 g0;
  g0.x = 1u;                                                   // count=1, user descriptor
  g0.y = lds_off;                                              // lds_addr [63:32]
  g0.z = (uint32_t)gaddr;                                      // global_addr[31:0]
  g0.w = ((uint32_t)(gaddr >> 32) & 0x01FFFFFFu) | (2u << 30); // global_addr[56:32] | type=2
  i32x8 g1;
  g1[0] = (int)(2u << 16);                                     // data_size=2 (4 bytes)
  g1[1] = (int)((cols & 0xFFFFu) << 16);                       // tensor_dim0 lo16 @ [63:48]
  g1[2] = (int)((cols >> 16) | ((rows & 0xFFFFu) << 16));      // tensor_dim0 hi | tensor_dim1 lo
  g1[3] = (int)((rows >> 16) | ((cols & 0xFFFFu) << 16));      // tensor_dim1 hi | tile_dim0
  g1[4] = (int)(rows & 0xFFFFu);                               // tile_dim1 | tile_dim2=0
  g1[5] = (int)row_stride_elems;                               // tensor_dim0_stride[31:0]
  g1[6] = 0;                                                   // stride hi | dim1_stride lo
  g1[7] = 0;
  i32x4 z4 = {0, 0, 0, 0};
#if __clang_major__ >= 23
  i32x8 z8 = {0, 0, 0, 0, 0, 0, 0, 0};
  __builtin_amdgcn_tensor_load_to_lds(g0, g1, z4, z4, z8, 0);
#else
  __builtin_amdgcn_tensor_load_to_lds(g0, g1, z4, z4, 0);
#endif
#else
  (void)lds_off; (void)gaddr; (void)rows; (void)cols; (void)row_stride_elems;
#endif
}

__global__ __launch_bounds__(256) void hbv_scan_kernel(
    const float* __restrict__ x,     // [T, G, 3] prcp, tmean, pet
    const float* __restrict__ praw,  // [G, 12, 8] raw params in [0,1)
    float* __restrict__ out,         // [T, G, 8]
    int G, int T) {
  __shared__ float sbuf[2][TCH * GPB * 3];

  const int tid   = threadIdx.x;
  const int m     = tid & (NMUL - 1);
  const int gl    = tid >> 3;
  const int gBase = blockIdx.x * GPB;
  int g = gBase + gl;
  const bool valid = (g < G);
  if (!valid) g = G - 1;

  const int gcells = min(GPB, G - gBase);   // cells covered by this block
  const int cols   = gcells * 3;            // floats per row in the LDS tile
  const int lbase  = valid ? gl * 3 : 0;

  // ---- Parameters: scale raw [0,1) into physical bounds ----
  const float* pg = praw + (size_t)g * 96 + m;
  const float BETA  = fmaf(pg[0],   5.0f,   1.0f);
  const float FC    = fmaf(pg[8],   950.0f, 50.0f);
  const float K0    = fmaf(pg[16],  0.85f,  0.05f);
  const float K1    = fmaf(pg[24],  0.49f,  0.01f);
  const float K2    = fmaf(pg[32],  0.199f, 0.001f);
  const float LP    = fmaf(pg[40],  0.8f,   0.2f);
  const float PERCc = pg[48] * 10.0f;
  const float UZL   = pg[56] * 100.0f;
  const float TTp   = fmaf(pg[64],  5.0f,  -2.5f);
  const float CFMAX = fmaf(pg[72],  9.5f,   0.5f);
  const float CFR   = pg[80] * 0.1f;
  const float CWH   = pg[88] * 0.2f;
  const float invFC    = 1.0f / FC;
  const float invLPFC  = 1.0f / (LP * FC);
  const float CFRCF    = CFR * CFMAX;

  // ---- State ----
  float SP  = 0.001f;   // SNOWPACK
  float MW  = 0.001f;   // MELTWATER
  float SM  = 0.001f;   // soil moisture
  float SUZ = 0.001f;
  float SLZ = 0.001f;

  const int nch = (T + TCH - 1) / TCH;
  const uint32_t rstride = (uint32_t)G * 3u;

  // ---- Prefetch chunk 0 ----
  {
    const int rows0 = min(TCH, T);
#if HAVE_TDM
    if (tid < 32) {
      tdm_load_tile_2d((uint32_t)(uintptr_t)&sbuf[0][0],
                       (uint64_t)(uintptr_t)(x + (size_t)gBase * 3),
                       (uint32_t)rows0, (uint32_t)cols, rstride);
      __builtin_amdgcn_s_wait_tensorcnt(0);
    }
#else
    const float* src0 = x + (size_t)gBase * 3;
    for (int i = tid; i < rows0 * cols; i += 256) {
      int r = i / cols, cc = i - r * cols;
      sbuf[0][r * cols + cc] = src0[(size_t)r * rstride + cc];
    }
#endif
    __syncthreads();
  }

  for (int c = 0; c < nch; ++c) {
    const int t0   = c * TCH;
    const int rows = min(TCH, T - t0);

    // ---- Async prefetch of next chunk into the other buffer ----
    if (c + 1 < nch) {
      const int rows1 = min(TCH, T - (t0 + TCH));
      const float* src = x + ((size_t)(t0 + TCH) * G + gBase) * 3;
#if HAVE_TDM
      if (tid < 32) {
        tdm_load_tile_2d((uint32_t)(uintptr_t)&sbuf[(c + 1) & 1][0],
                         (uint64_t)(uintptr_t)src,
                         (uint32_t)rows1, (uint32_t)cols, rstride);
      }
#else
      float* dst = &sbuf[(c + 1) & 1][0];
      for (int i = tid; i < rows1 * cols; i += 256) {
        int r = i / cols, cc = i - r * cols;
        dst[r * cols + cc] = src[(size_t)r * rstride + cc];
      }
#endif
    }

    // ---- Process current chunk from LDS ----
    const float* sb = &sbuf[c & 1][0];
    float* op = out + ((size_t)t0 * G + g) * NMUL + m;
    const size_t ostride = (size_t)G * NMUL;

    for (int r = 0; r < rows; ++r) {
      const int b = r * cols + lbase;
      const float P   = sb[b + 0];
      const float Tt  = sb[b + 1];
      const float PET = sb[b + 2];

      // Snow routine
      const float rain = (Tt >= TTp) ? P : 0.0f;
      const float snow = P - rain;
      SP += snow;
      const float melt = fminf(fmaxf(CFMAX * (Tt - TTp), 0.0f), SP);
      MW += melt;  SP -= melt;
      const float refr = fminf(fmaxf(CFRCF * (TTp - Tt), 0.0f), MW);
      SP += refr;  MW -= refr;
      const float tosoil = fmaxf(MW - CWH * SP, 0.0f);
      MW -= tosoil;

      // Soil routine: (SM/FC)^BETA via v_log_f32/v_exp_f32 (SM >= 1e-5 > 0)
      float sw = fast_exp2(BETA * fast_log2(SM * invFC));
      sw = fminf(sw, 1.0f);
      const float inflow   = rain + tosoil;
      const float recharge = inflow * sw;
      SM += inflow - recharge;
      const float excess = fmaxf(SM - FC, 0.0f);
      SM -= excess;
      const float evap = fminf(SM * invLPFC, 1.0f);
      const float ET   = fminf(SM, PET * evap);
      SM = fmaxf(SM - ET, 1e-5f);

      // Response routine
      SUZ += recharge + excess;
      const float PERC = fminf(SUZ, PERCc);
      SUZ -= PERC;
      const float Q0 = K0 * fmaxf(SUZ - UZL, 0.0f);
      SUZ -= Q0;
      const float Q1 = K1 * SUZ;
      SUZ -= Q1;
      SLZ += PERC;
      const float Q2 = K2 * SLZ;
      SLZ -= Q2;

      if (valid) __builtin_nontemporal_store(Q0 + Q1 + Q2, op);
      op += ostride;
    }

#if HAVE_TDM
    if (tid < 32) __builtin_amdgcn_s_wait_tensorcnt(0);
#endif
    __syncthreads();
  }
}

extern "C" void kernel_launch(void* const* d_in, const int* in_sizes, int n_in,
                              void* d_out, int out_size, void* d_ws, size_t ws_size,
                              hipStream_t stream) {
  (void)n_in; (void)out_size; (void)d_ws; (void)ws_size;
  const float* x    = (const float*)d_in[0];   // [T, G, 3] f32
  const float* praw = (const float*)d_in[1];   // [1, G, 12, 8] f32
  float* out = (float*)d_out;                  // [T, G, 8] f32

  const int G = in_sizes[1] / (12 * NMUL);     // 4000
  const int T = in_sizes[0] / (3 * G);         // 730

  const int blocks = (G + GPB - 1) / GPB;
  hbv_scan_kernel<<<blocks, 256, 0, stream>>>(x, praw, out, G, T);
}